// BlurLayer_exp_76338748719473
// MI455X (gfx1250) — compile-verified
//
#include <hip/hip_runtime.h>
#include <math.h>

// ---------------------------------------------------------------------------
// Ideal square low-pass filter == separable circular convolution with the
// Dirichlet kernel:  out_u = K_u * X * K_u,  K_u[i][j] = h_u((i-j) mod 256).
// All GEMMs run on v_wmma_f32_16x16x4_f32 (full fp32 precision).
// h tables are stored in LDS as pre-paired float2 (forward and reversed), so
// every WMMA A/B operand fragment is ONE aligned ds_load_b64 whose address is
// linear in k -> immediate offsets, no per-iteration address VALU.
// ---------------------------------------------------------------------------

typedef __attribute__((ext_vector_type(2))) float v2f;
typedef __attribute__((ext_vector_type(8))) float v8f;

#define NPIX 256
#define NU 8
#define NCH (NU + 1)
#define TSTRIDE 260   // 256 + 4: even, lane stride 4 banks -> conflict-free
#define HRLEN 516     // reversed pair-table length (max index 510)
#define HBLEN 512     // forward pair-table length (max index 510)

__constant__ int c_us[NU] = {5, 22, 39, 57, 74, 92, 109, 127};

// --- h_u tables: h_u(d) = sin(pi*(2u+1)*d/N) / (N*sin(pi*d/N)) --------------
__global__ void dirichlet_tables(float* __restrict__ htab) {
  const int u = blockIdx.x;     // 0..7
  const int d = threadIdx.x;    // 0..255
  const int m = 2 * c_us[u] + 1;
  float v;
  if (d == 0) {
    v = (float)m / (float)NPIX;
  } else {
    const double pd = 3.14159265358979323846 * (double)d / (double)NPIX;
    v = (float)(sin((double)m * pd) / ((double)NPIX * sin(pd)));
  }
  htab[u * NPIX + d] = v;
}

// --- channel 0 is the input itself ------------------------------------------
__global__ void copy_ch0(const float* __restrict__ x, float* __restrict__ out) {
  const size_t gid = (size_t)blockIdx.x * blockDim.x + threadIdx.x;
  out[gid * NCH] = x[gid];
}

// --- fused strip GEMM: O[strip] = K_u[strip rows] @ X_b @ K_u ---------------
__global__ __launch_bounds__(128) void lowpass_gemm(
    const float* __restrict__ x,     // [b, 256, 256]
    const float* __restrict__ htab,  // [8, 256]
    float* __restrict__ out) {       // [b, 256, 256, 9]
  __shared__ v2f hrp[HRLEN];             // hrp[j] = { h[(256-j)&255], h[(255-j)&255] }
  __shared__ v2f hbp[HBLEN];             // hbp[j] = { h[j&255],       h[(j+1)&255]   }
  __shared__ float t_lds[16 * TSTRIDE];

  const int strip = blockIdx.x;  // 0..15 (16-row output strip)
  const int u     = blockIdx.y;  // 0..7
  const int b     = blockIdx.z;  // batch

  const int tid  = threadIdx.x;
  const int wave = tid >> 5;
  const int lane = tid & 31;
  const int half = lane >> 4;    // which 16-lane half (K split / M split)
  const int l    = lane & 15;

  {
    const float* __restrict__ hu = htab + u * NPIX;
    for (int i = tid; i < HRLEN; i += 128) {
      v2f p;
      p.x = hu[(NPIX - i) & 255];
      p.y = hu[(NPIX - 1 - i) & 255];
      hrp[i] = p;
    }
    for (int i = tid; i < HBLEN; i += 128) {
      v2f p;
      p.x = hu[i & 255];
      p.y = hu[(i + 1) & 255];
      hbp[i] = p;
    }
  }
  __syncthreads();

  const float* __restrict__ xb = x + (size_t)b * (NPIX * NPIX);
  const int mi16 = strip * 16;
  const int h2   = 2 * half;

  // GEMM1 A fragment: { h[(mi16+l-h2-k)&255], h[(mi16+l-h2-k-1)&255] }
  //   = hrp[256 - (mi16+l-h2) + k]  -> one aligned b64 load, linear in k
  const v2f* __restrict__ ha = &hrp[256 - mi16 - l + h2];
  // GEMM1 B (global): X[k+h2+j][n]
  const float* __restrict__ xk = xb + (size_t)h2 * NPIX;
  // GEMM2 A (lds): T[l][k+h2 .. +1]; index always even -> 8B aligned
  const float* __restrict__ ta = &t_lds[l * TSTRIDE + h2];

  // ---- GEMM 1: T(16x256) = K_u[mi16..mi16+15][:] @ X_b ----
  // wave w owns columns [w*64, w*64+64), 4 tiles of 16
  for (int tt = 0; tt < 4; ++tt) {
    const int colbase = (wave * 4 + tt) * 16;
    const int n = colbase + l;
    v8f acc = {0.f, 0.f, 0.f, 0.f, 0.f, 0.f, 0.f, 0.f};
#pragma unroll 8
    for (int k = 0; k < NPIX; k += 4) {
      v2f a = ha[k];
      v2f bm;
      bm.x = xk[(size_t)k * NPIX + n];
      bm.y = xk[(size_t)k * NPIX + NPIX + n];
      acc = __builtin_amdgcn_wmma_f32_16x16x4_f32(
          false, a, false, bm, (short)0, acc, false, false);
    }
    // C/D layout: VGPR r holds rows r (lanes 0-15) and r+8 (lanes 16-31)
#pragma unroll
    for (int r = 0; r < 8; ++r)
      t_lds[(r + 8 * half) * TSTRIDE + n] = acc[r];
  }
  __syncthreads();

  // ---- GEMM 2: O(16x256) = T @ K_u ----
  for (int tt = 0; tt < 4; ++tt) {
    const int colbase = (wave * 4 + tt) * 16;
    const int n = colbase + l;
    // B fragment: { h[(k+h2-n)&255], h[(k+h2+1-n)&255] } = hbp[256 - n + h2 + k]
    const v2f* __restrict__ hb = &hbp[256 - n + h2];
    v8f acc = {0.f, 0.f, 0.f, 0.f, 0.f, 0.f, 0.f, 0.f};
#pragma unroll 8
    for (int k = 0; k < NPIX; k += 4) {
      v2f a = *(const v2f*)&ta[k];   // aligned ds_load_b64
      v2f bm = hb[k];                // aligned ds_load_b64
      acc = __builtin_amdgcn_wmma_f32_16x16x4_f32(
          false, a, false, bm, (short)0, acc, false, false);
    }
#pragma unroll
    for (int r = 0; r < 8; ++r) {
      const int i = mi16 + r + 8 * half;
      out[(((size_t)b * NPIX + i) * NPIX + n) * NCH + 1 + u] = acc[r];
    }
  }
}

extern "C" void kernel_launch(void* const* d_in, const int* in_sizes, int n_in,
                              void* d_out, int out_size, void* d_ws, size_t ws_size,
                              hipStream_t stream) {
  const float* x = (const float*)d_in[0];
  float* out = (float*)d_out;
  float* htab = (float*)d_ws;  // 8 * 256 floats = 8 KB scratch

  const int nbatch = in_sizes[0] / (NPIX * NPIX);  // 64
  const size_t npix_total = (size_t)nbatch * NPIX * NPIX;

  dirichlet_tables<<<dim3(NU), dim3(NPIX), 0, stream>>>(htab);
  copy_ch0<<<dim3((unsigned)(npix_total / 256)), dim3(256), 0, stream>>>(x, out);
  lowpass_gemm<<<dim3(16, NU, nbatch), dim3(128), 0, stream>>>(x, htab, out);
}